// Adaptor_39960375722661
// MI455X (gfx1250) — compile-verified
//
#include <hip/hip_runtime.h>
#include <math.h>

typedef float v2f __attribute__((ext_vector_type(2)));
typedef float v8f __attribute__((ext_vector_type(8)));

#define EPS 1e-5f

__device__ __forceinline__ float silu_f(float x) { return x / (1.0f + __expf(-x)); }

// im2col offset helpers: called only with compile-time-constant k (after full unroll),
// so ic/ky/kx fold to constants and the offset becomes  const + sr18[ky] + sc[kx].
__device__ __forceinline__ int a_off36(int k, const int r18[3], const int c[3]) {
    int ic = k / 9, t = k - ic * 9, ky = t / 3, kx = t - ky * 3;
    return ic * 324 + r18[ky] + c[kx];
}

// ---------------------------------------------------------------- K1: 4x4 avg pool
// x (16,320,64,64) -> z0 (16,320,16,16).  Reads the 84MB input (dominant BW).
__global__ void k_pool(const float* __restrict__ x, float* __restrict__ z0) {
    int o = blockIdx.x * 256 + threadIdx.x;        // < 1310720
    int xc = o & 15;
    int rest = o >> 4;
    int yr = rest & 15;
    int bc = rest >> 4;                            // b*320 + c
    const float* src = x + ((size_t)bc * 64 + (size_t)yr * 4) * 64 + (size_t)xc * 4;
    float s = 0.f;
#pragma unroll
    for (int r = 0; r < 4; r++) {
        float4 v = *reinterpret_cast<const float4*>(src + (size_t)r * 64);
        s += v.x + v.y + v.z + v.w;
    }
    z0[o] = s * 0.0625f;
}

// ---------------------------------------------------------------- K2: GN1 stats (32 groups of 10 ch)
__global__ void k_gn1_stats(const float* __restrict__ z0, float* __restrict__ st) {
    int bg = blockIdx.x;                           // b*32 + g  (512 blocks)
    const float* p = z0 + (size_t)bg * 2560;       // 10 ch * 256 spatial, contiguous
    float s = 0.f, q = 0.f;
    for (int i = threadIdx.x; i < 2560; i += 256) { float v = p[i]; s += v; q += v * v; }
    __shared__ float ss[256], sq[256];
    ss[threadIdx.x] = s; sq[threadIdx.x] = q;
    __syncthreads();
    for (int o = 128; o > 0; o >>= 1) {
        if (threadIdx.x < o) { ss[threadIdx.x] += ss[threadIdx.x + o]; sq[threadIdx.x] += sq[threadIdx.x + o]; }
        __syncthreads();
    }
    if (threadIdx.x == 0) {
        float m = ss[0] * (1.f / 2560.f);
        float var = sq[0] * (1.f / 2560.f) - m * m;
        st[bg * 2] = m;
        st[bg * 2 + 1] = rsqrtf(var + EPS);
    }
}

// ---------------------------------------------------------------- K3: GN1+SiLU -> zero-halo padded 18x18
__global__ void k_act1(const float* __restrict__ z0, const float* __restrict__ st,
                       const float* __restrict__ g1, const float* __restrict__ b1,
                       float* __restrict__ snp) {
    unsigned idx = blockIdx.x * 256u + threadIdx.x;  // < 1658880 = 16*320*324
    unsigned p = idx % 324u;
    unsigned bc = idx / 324u;                        // b*320 + c
    unsigned py = p / 18u, px = p % 18u;
    float out = 0.f;
    if (py >= 1u && py <= 16u && px >= 1u && px <= 16u) {
        unsigned c = bc % 320u, b = bc / 320u;
        float v = z0[bc * 256u + (py - 1u) * 16u + (px - 1u)];
        unsigned g = c / 10u;
        float m = st[(b * 32u + g) * 2u];
        float rs = st[(b * 32u + g) * 2u + 1u];
        float xn = (v - m) * rs * g1[c] + b1[c];
        out = silu_f(xn);
    }
    snp[idx] = out;
}

// ---------------------------------------------------------------- K4: conv1 via WMMA f32 16x16x4
// im2col GEMM:  M = b*256 + y*16 + x (4096),  N = 8 oc (padded to 16),  K = 320*9 = 2880.
// K-loop runs in blocks of 36 (= 4 ic x 9 taps) so the tap decomposition is compile-time;
// the khalf-dependent k offset is handled by computing both constant candidates + cndmask.
__global__ void k_conv1(const float* __restrict__ snp, const float* __restrict__ w1,
                        const float* __restrict__ bias1, float* __restrict__ z1) {
    int lane = threadIdx.x & 31;
    int wv = threadIdx.x >> 5;
    int mtile = blockIdx.x * 8 + wv;               // 0..255
    int mbase = mtile * 16;
    int b = mbase >> 8;                            // tile never crosses batch (16 | 256)
    int mrow = lane & 15;
    int khalf = lane >> 4;
    int m = mbase + mrow;
    int yx = m & 255;
    int y = yx >> 4, xx = yx & 15;
    int ncol = lane & 15;
    const float* abase = snp + (size_t)b * 320 * 324;
    const float* wrow = w1 + (size_t)(ncol & 7) * 2880;  // in-range even for pad lanes
    bool nok = (ncol < 8);

    int r18[3], cx[3];
#pragma unroll
    for (int t = 0; t < 3; t++) { r18[t] = (y + t) * 18; cx[t] = xx + t; }  // halo +1 folded in

    v8f acc = {0.f, 0.f, 0.f, 0.f, 0.f, 0.f, 0.f, 0.f};
    for (int ic4 = 0; ic4 < 80; ic4++) {           // 80 blocks of 4 input channels
        const float* ab4 = abase + ic4 * 1296;
        const float* wb4 = wrow + ic4 * 36;
#pragma unroll
        for (int j = 0; j < 9; j++) {
            v2f a, bb;
#pragma unroll
            for (int e = 0; e < 2; e++) {
                int ja = j * 4 + e;                // khalf==0 candidate (compile-time)
                int jb = ja + 2;                   // khalf==1 candidate (compile-time)
                int offA = a_off36(ja, r18, cx);
                int offB = a_off36(jb, r18, cx);
                a[e] = ab4[khalf ? offB : offA];
                float wvv = wb4[khalf ? jb : ja];
                bb[e] = nok ? wvv : 0.0f;          // pad N to 16 with zeros
            }
            acc = __builtin_amdgcn_wmma_f32_16x16x4_f32(false, a, false, bb, (short)0, acc, false, false);
        }
    }
    if (nok) {
        float bv = bias1[ncol];
#pragma unroll
        for (int j = 0; j < 8; j++) {
            int mm = mbase + j + 8 * khalf;
            z1[(((size_t)b * 8 + (size_t)ncol) << 8) + (size_t)(mm & 255)] = acc[j] + bv;
        }
    }
}

// ---------------------------------------------------------------- K5: GN_a + QKV + attention + proj + LN + MLP + GN2/SiLU
// One workgroup (8 wave32) per batch.  Writes padded activated pre-upsample source (16,8,18,18).
__global__ void k_mid(const float* __restrict__ z1g,
                      const float* __restrict__ ga, const float* __restrict__ ba,
                      const float* __restrict__ wqkv, const float* __restrict__ bqkv,
                      const float* __restrict__ wproj, const float* __restrict__ bproj,
                      const float* __restrict__ lng, const float* __restrict__ lnb,
                      const float* __restrict__ wfc1, const float* __restrict__ bfc1,
                      const float* __restrict__ wfc2, const float* __restrict__ bfc2,
                      const float* __restrict__ g2, const float* __restrict__ b2,
                      float* __restrict__ spad2) {
    __shared__ float z1s[8][256];
    __shared__ float qk[24][256];
    __shared__ float zs[8][256];
    __shared__ float wq[24][8];
    __shared__ float bq[24];
    __shared__ float mA[8], rA[8], m2[8], r2[8];
    int b = blockIdx.x;
    int tid = threadIdx.x, lane = tid & 31, wv = tid >> 5;

    const float* zp = z1g + (size_t)b * 2048;
    for (int i = tid; i < 2048; i += 256) z1s[i >> 8][i & 255] = zp[i];
    for (int i = tid; i < 192; i += 256) wq[i >> 3][i & 7] = wqkv[i];
    if (tid < 24) bq[tid] = bqkv[tid];
    __syncthreads();

    // per-channel GN_a stats: wave wv reduces channel wv over 256 elems
    {
        float s = 0.f, q = 0.f;
        for (int i = lane; i < 256; i += 32) { float v = z1s[wv][i]; s += v; q += v * v; }
        for (int o = 16; o > 0; o >>= 1) { s += __shfl_xor(s, o, 32); q += __shfl_xor(q, o, 32); }
        if (lane == 0) {
            float m = s * (1.f / 256.f);
            float var = q * (1.f / 256.f) - m * m;
            mA[wv] = m; rA[wv] = rsqrtf(var + EPS);
        }
    }
    __syncthreads();

    int n = tid;                                    // one thread per spatial position
    float av[8];
#pragma unroll
    for (int c = 0; c < 8; c++) av[c] = (z1s[c][n] - mA[c]) * rA[c] * ga[c] + ba[c];
#pragma unroll
    for (int j = 0; j < 24; j++) {
        float s = bq[j];
#pragma unroll
        for (int c = 0; c < 8; c++) s += wq[j][c] * av[c];
        qk[j][n] = s;
    }
    __syncthreads();

    // attention: 4 heads, dh=2, seq 256.  LDS-broadcast loops over m.
    float att[8];
    const float scale = 0.70710678118654752f;       // dh^-0.5
#pragma unroll
    for (int h = 0; h < 4; h++) {
        float q0 = qk[0 + h * 2][n], q1 = qk[1 + h * 2][n];
        const float* k0 = qk[8 + h * 2];
        const float* k1 = qk[9 + h * 2];
        const float* v0 = qk[16 + h * 2];
        const float* v1 = qk[17 + h * 2];
        float mx = -1e30f;
        for (int mi = 0; mi < 256; mi++) {
            float s = (q0 * k0[mi] + q1 * k1[mi]) * scale;
            mx = fmaxf(mx, s);
        }
        float se = 0.f, o0 = 0.f, o1 = 0.f;
        for (int mi = 0; mi < 256; mi++) {
            float s = (q0 * k0[mi] + q1 * k1[mi]) * scale;
            float p = __expf(s - mx);
            se += p; o0 += p * v0[mi]; o1 += p * v1[mi];
        }
        float inv = 1.0f / se;
        att[h * 2 + 0] = o0 * inv;
        att[h * 2 + 1] = o1 * inv;
    }

    // proj + residual
    float r[8];
#pragma unroll
    for (int c = 0; c < 8; c++) {
        float s = bproj[c];
#pragma unroll
        for (int cc = 0; cc < 8; cc++) s += wproj[c * 8 + cc] * att[cc];
        r[c] = z1s[c][n] + s;
    }

    // LayerNorm over 8 channels
    float mu = 0.f;
#pragma unroll
    for (int c = 0; c < 8; c++) mu += r[c];
    mu *= 0.125f;
    float var = 0.f;
#pragma unroll
    for (int c = 0; c < 8; c++) { float d = r[c] - mu; var += d * d; }
    var *= 0.125f;
    float rs = rsqrtf(var + EPS);
    float ln[8];
#pragma unroll
    for (int c = 0; c < 8; c++) ln[c] = (r[c] - mu) * rs * lng[c] + lnb[c];

    // MLP: fc1 (16x8) + exact GELU + fc2 (8x16) + residual
    float hg[16];
#pragma unroll
    for (int j = 0; j < 16; j++) {
        float s = bfc1[j];
#pragma unroll
        for (int c = 0; c < 8; c++) s += wfc1[j * 8 + c] * ln[c];
        hg[j] = 0.5f * s * (1.0f + erff(s * 0.70710678f));
    }
    float zt[8];
#pragma unroll
    for (int c = 0; c < 8; c++) {
        float s = bfc2[c];
#pragma unroll
        for (int j = 0; j < 16; j++) s += wfc2[c * 16 + j] * hg[j];
        zt[c] = r[c] + s;
        zs[c][n] = zt[c];
    }
    __syncthreads();

    // GN2 per-channel stats (stats over 16x16 == stats over 4x-upsampled 64x64)
    {
        float s = 0.f, q = 0.f;
        for (int i = lane; i < 256; i += 32) { float v = zs[wv][i]; s += v; q += v * v; }
        for (int o = 16; o > 0; o >>= 1) { s += __shfl_xor(s, o, 32); q += __shfl_xor(q, o, 32); }
        if (lane == 0) {
            float m = s * (1.f / 256.f);
            float vv = q * (1.f / 256.f) - m * m;
            m2[wv] = m; r2[wv] = rsqrtf(vv + EPS);
        }
    }
    __syncthreads();

    // GN2 + SiLU into zero-halo padded (8,18,18) pre-upsample source for conv2
    float* op = spad2 + (size_t)b * 8 * 324;
    for (int i = tid; i < 2592; i += 256) {
        int c = i / 324, p = i % 324;
        int py = p / 18, px = p % 18;
        float out = 0.f;
        if (py >= 1 && py <= 16 && px >= 1 && px <= 16) {
            float v = zs[c][(py - 1) * 16 + (px - 1)];
            float xn = (v - m2[c]) * r2[c] * g2[c] + b2[c];
            out = silu_f(xn);
        }
        op[i] = out;
    }
}

// ---------------------------------------------------------------- K6: conv2 via WMMA f32 16x16x4
// GEMM: M = b*4096 + y*64 + x (65536), N = 320 oc, K = 8*9 = 72.
// Each wave keeps 5 accumulators (80 oc) so one A fragment feeds 5 WMMAs per k-step
// (A im2col build amortized 5x).  Upsample folded into addressing via ((y+ky-1)>>2)+1
// into the zero-halo'd 18x18 source.  LDS phase-aliased: B panel (80x72) during the
// k-loop, then the 80x128 D stage, written back as float4 rows (512B contiguous).
__global__ void k_conv2(const float* __restrict__ spad2, const float* __restrict__ w2,
                        const float* __restrict__ bias2, float* __restrict__ out) {
    __shared__ float smem[10240];          // max(80*72, 80*128) floats = 40KB
    int ng = blockIdx.x;                   // 0..3  -> 80 output channels each
    int mb = blockIdx.y;                   // 0..511
    int tid = threadIdx.x, lane = tid & 31, wv = tid >> 5;
    int oc0 = ng * 80;

    // phase 1: B panel, contiguous slice of w_conv2 (oc-major, 72 floats/oc)
    for (int i = tid; i < 5760; i += 256) smem[i] = w2[(size_t)oc0 * 72 + i];
    __syncthreads();

    int mtbase = mb * 128 + wv * 16;
    int b = mtbase >> 12;                  // 128 | 4096: block never crosses batch
    int mrow = lane & 15;
    int khalf = lane >> 4;
    int m = mtbase + mrow;
    int yx = m & 4095;
    int y = yx >> 6, xx = yx & 63;
    int ncol = lane & 15;
    const float* ab = spad2 + (size_t)b * 8 * 324;

    int r18[3], cx[3];
#pragma unroll
    for (int t = 0; t < 3; t++) {
        r18[t] = ((((y + t - 1) >> 2) + 1) * 18);   // arithmetic shift: -1 -> halo row 0
        cx[t]  = (((xx + t - 1) >> 2) + 1);
    }

    v8f zero = {0.f, 0.f, 0.f, 0.f, 0.f, 0.f, 0.f, 0.f};
    v8f acc[5];
#pragma unroll
    for (int tt = 0; tt < 5; tt++) acc[tt] = zero;

#pragma unroll
    for (int kk = 0; kk < 72; kk += 4) {
        v2f a;
#pragma unroll
        for (int e = 0; e < 2; e++) {
            int offA = a_off36(kk + e,     r18, cx);  // khalf==0 candidate (const-folded)
            int offB = a_off36(kk + e + 2, r18, cx);  // khalf==1 candidate (const-folded)
            a[e] = ab[khalf ? offB : offA];
        }
#pragma unroll
        for (int tt = 0; tt < 5; tt++) {
            const float* br = &smem[(tt * 16 + ncol) * 72 + kk + khalf * 2];
            v2f bb;
            bb[0] = br[0];
            bb[1] = br[1];
            acc[tt] = __builtin_amdgcn_wmma_f32_16x16x4_f32(false, a, false, bb, (short)0, acc[tt], false, false);
        }
    }
    __syncthreads();                       // all waves done reading B panel

    // phase 2: stage D tiles as [oc 80][m 128]
#pragma unroll
    for (int tt = 0; tt < 5; tt++) {
#pragma unroll
        for (int j = 0; j < 8; j++) {
            int mloc = wv * 16 + j + 8 * khalf;
            smem[(tt * 16 + ncol) * 128 + mloc] = acc[tt][j];
        }
    }
    __syncthreads();

    // coalesced float4 writeback; each oc row is 128 contiguous floats in out
    size_t outbase = ((size_t)b * 320 + (size_t)oc0) * 4096 + (size_t)(mb * 128 & 4095);
    for (int i = tid; i < 2560; i += 256) {
        int nn = i >> 5;
        int mm4 = (i & 31) << 2;
        float4 v = *reinterpret_cast<const float4*>(&smem[nn * 128 + mm4]);
        float bv = bias2[oc0 + nn];
        v.x += bv; v.y += bv; v.z += bv; v.w += bv;
        *reinterpret_cast<float4*>(&out[outbase + (size_t)nn * 4096 + (size_t)mm4]) = v;
    }
}

// ---------------------------------------------------------------- launch
extern "C" void kernel_launch(void* const* d_in, const int* in_sizes, int n_in,
                              void* d_out, int out_size, void* d_ws, size_t ws_size,
                              hipStream_t stream) {
    const float* x       = (const float*)d_in[0];
    const float* g1      = (const float*)d_in[1];
    const float* b1      = (const float*)d_in[2];
    const float* w_conv1 = (const float*)d_in[3];
    const float* b_conv1 = (const float*)d_in[4];
    const float* ga      = (const float*)d_in[5];
    const float* ba      = (const float*)d_in[6];
    const float* w_qkv   = (const float*)d_in[7];
    const float* b_qkv   = (const float*)d_in[8];
    const float* w_proj  = (const float*)d_in[9];
    const float* b_proj  = (const float*)d_in[10];
    const float* ln_g    = (const float*)d_in[11];
    const float* ln_b    = (const float*)d_in[12];
    const float* w_fc1   = (const float*)d_in[13];
    const float* b_fc1   = (const float*)d_in[14];
    const float* w_fc2   = (const float*)d_in[15];
    const float* b_fc2   = (const float*)d_in[16];
    const float* g2      = (const float*)d_in[17];
    const float* b2      = (const float*)d_in[18];
    const float* w_conv2 = (const float*)d_in[19];
    const float* b_conv2 = (const float*)d_in[20];
    float* out = (float*)d_out;

    float* ws  = (float*)d_ws;
    float* z0  = ws;                  // 1,310,720
    float* st1 = z0  + 1310720;       // 1,024
    float* snp = st1 + 1024;          // 1,658,880
    float* z1  = snp + 1658880;       // 32,768
    float* sp2 = z1  + 32768;         // 41,472   (total ~12.2 MB)

    k_pool<<<5120, 256, 0, stream>>>(x, z0);
    k_gn1_stats<<<512, 256, 0, stream>>>(z0, st1);
    k_act1<<<6480, 256, 0, stream>>>(z0, st1, g1, b1, snp);
    k_conv1<<<32, 256, 0, stream>>>(snp, w_conv1, b_conv1, z1);
    k_mid<<<16, 256, 0, stream>>>(z1, ga, ba, w_qkv, b_qkv, w_proj, b_proj,
                                  ln_g, ln_b, w_fc1, b_fc1, w_fc2, b_fc2, g2, b2, sp2);
    k_conv2<<<dim3(4, 512), 256, 0, stream>>>(sp2, w_conv2, b_conv2, out);
}